// PedFFNN_Cross_13718125543489
// MI455X (gfx1250) — compile-verified
//
#include <hip/hip_runtime.h>

typedef __attribute__((ext_vector_type(2))) float v2f;
typedef __attribute__((ext_vector_type(4))) float v4f;
typedef __attribute__((ext_vector_type(8))) float v8f;

#define B_ROWS 65536
#define F_DIM  512
#define T_DIM  8
#define L_CROSS 4
#define KBLKS  (F_DIM / 4)     // 128 WMMA k-steps of 4
#define NPAD   16              // 4 cross dots + 8 out cols + 4 zero pad

// ---------------------------------------------------------------------------
// Kernel 1: tiny precompute of the collapsed-recurrence scalars.
//   cs[i] = sum_{j<i} cross_b[j] . cross_w[i]            (4 floats)
//   P[t]  = (sum_j cross_b[j]) . W_out[:,t] + b_out[t]   (8 floats)
// One block, 24 active waves (16 for the c-matrix, 8 for P).
// ---------------------------------------------------------------------------
__global__ __launch_bounds__(768) void precompute_kernel(
    const float* __restrict__ cross_w, const float* __restrict__ cross_b,
    const float* __restrict__ W_out, const float* __restrict__ b_out,
    float* __restrict__ csP /* [12]: cs[4], P[8] */) {
  __shared__ float cmat[16];
  const int tid = threadIdx.x;
  const int wave = tid >> 5;
  const int lane = tid & 31;

  if (wave < 16) {  // c[j][i] = b_j . w_i
    const int j = wave >> 2, i = wave & 3;
    float acc = 0.f;
    for (int f = lane; f < F_DIM; f += 32)
      acc += cross_b[j * F_DIM + f] * cross_w[i * F_DIM + f];
    for (int off = 16; off > 0; off >>= 1) acc += __shfl_xor(acc, off, 32);
    if (lane == 0) cmat[wave] = acc;
  }

  float pacc = 0.f;
  const int t = wave - 16;
  if (wave >= 16 && wave < 24) {  // P[t]
    for (int f = lane; f < F_DIM; f += 32) {
      float bs = cross_b[f] + cross_b[F_DIM + f] +
                 cross_b[2 * F_DIM + f] + cross_b[3 * F_DIM + f];
      pacc += bs * W_out[f * T_DIM + t];
    }
    for (int off = 16; off > 0; off >>= 1) pacc += __shfl_xor(pacc, off, 32);
    if (lane == 0) pacc += b_out[t];
  }

  __syncthreads();

  if (tid < 4) {  // cs[i] = sum_{j<i} c[j][i]
    float s = 0.f;
    for (int j = 0; j < tid; ++j) s += cmat[j * 4 + tid];
    csP[tid] = s;
  }
  if (wave >= 16 && wave < 24 && lane == 0) csP[4 + t] = pacc;
}

// ---------------------------------------------------------------------------
// Kernel 2: pack the 512x16 B matrix [w0..w3 | W_out | 0] into the per-lane
// WMMA f32 4x16 B-fragment layout:
//   VGPR0/1, lanes 0-15: K = kb*4 + {0,1}, N = lane
//            lanes 16-31: K = kb*4 + {2,3}, N = lane-16
// bpack[kb][lane][p], p in {0,1}  -> 128*32*2 floats = 32 KB.
// ---------------------------------------------------------------------------
__global__ __launch_bounds__(256) void pack_b_kernel(
    const float* __restrict__ cross_w, const float* __restrict__ W_out,
    float* __restrict__ bpack) {
  const int idx = blockIdx.x * 256 + threadIdx.x;  // 0 .. 8191
  if (idx >= KBLKS * 64) return;
  const int kblk = idx >> 6;
  const int rem = idx & 63;
  const int lane = rem >> 1;
  const int p = rem & 1;
  const int hi = lane >> 4;
  const int n = lane & 15;
  const int k = kblk * 4 + 2 * hi + p;
  float v = 0.f;
  if (n < L_CROSS)
    v = cross_w[n * F_DIM + k];
  else if (n < L_CROSS + T_DIM)
    v = W_out[k * T_DIM + (n - L_CROSS)];
  bpack[idx] = v;
}

// ---------------------------------------------------------------------------
// Kernel 3: main streaming pass. One wave per 16-row tile of X.
// D(16x16) = X_tile(16x512) x M(512x16) via 128x V_WMMA_F32_16X16X4_F32,
// then per-row alpha recurrence + epilogue.
// ---------------------------------------------------------------------------
__global__ __launch_bounds__(256) void dcn_wmma_kernel(
    const float* __restrict__ x, const float* __restrict__ bpack,
    const float* __restrict__ csP, float* __restrict__ out) {
  __shared__ float xfer[8][16 * 17];  // per-wave D-tile transpose buffer

  const int wave = threadIdx.x >> 5;
  const int lane = threadIdx.x & 31;
  const int tile = blockIdx.x * 8 + wave;  // 4096 tiles total
  const int row0 = tile * 16;

  const int hi = lane >> 4;       // 0: K even pair, 1: K odd pair
  const int mrow = lane & 15;     // A-matrix row for this lane

  // A fragment source: lane reads X[row0+mrow][kb*4 + 2*hi .. +1] (8B, aligned)
  const float* __restrict__ xrow =
      x + (size_t)(row0 + mrow) * F_DIM + 2 * hi;
  const float* __restrict__ bp = bpack + lane * 2;

  v8f acc0 = {};
  v8f acc1 = {};

#pragma unroll 4
  for (int kb = 0; kb < KBLKS; kb += 2) {
    v2f a0 = *(const v2f*)(xrow + kb * 4);
    v2f b0 = *(const v2f*)(bp + kb * 64);
    v2f a1 = *(const v2f*)(xrow + (kb + 1) * 4);
    v2f b1 = *(const v2f*)(bp + (kb + 1) * 64);
    // D = A x B + C, fp32, 16x16x4
    acc0 = __builtin_amdgcn_wmma_f32_16x16x4_f32(
        false, a0, false, b0, (short)0, acc0, false, false);
    acc1 = __builtin_amdgcn_wmma_f32_16x16x4_f32(
        false, a1, false, b1, (short)0, acc1, false, false);
  }
  v8f acc = acc0 + acc1;

  // --- transpose D tile through LDS ------------------------------------
  // C/D layout: lanes 0-15 = N=lane rows M=0..7 (vgpr r), lanes 16-31 =
  // N=lane-16 rows M=8..15.
  float* l = &xfer[wave][0];
  const int col = lane & 15;
#pragma unroll
  for (int r = 0; r < 8; ++r) l[(r + 8 * hi) * 17 + col] = acc[r];

  __syncthreads();

  // --- per-row collapsed recurrence + store ----------------------------
  if (lane < 16) {
    const float* rowd = l + lane * 17;  // d[0..3], u[0..7] for this row
    const float d0 = rowd[0], d1 = rowd[1], d2 = rowd[2], d3 = rowd[3];

    const float cs0 = csP[0], cs1 = csP[1], cs2 = csP[2], cs3 = csP[3];
    float alpha = 1.f;
    alpha += alpha * d0 + cs0;
    alpha += alpha * d1 + cs1;
    alpha += alpha * d2 + cs2;
    alpha += alpha * d3 + cs3;

    v4f o0, o1;
#pragma unroll
    for (int t = 0; t < 4; ++t) o0[t] = alpha * rowd[4 + t] + csP[4 + t];
#pragma unroll
    for (int t = 0; t < 4; ++t) o1[t] = alpha * rowd[8 + t] + csP[8 + t];

    float* orow = out + (size_t)(row0 + lane) * T_DIM;
    *(v4f*)(orow) = o0;
    *(v4f*)(orow + 4) = o1;
  }
}

// ---------------------------------------------------------------------------
extern "C" void kernel_launch(void* const* d_in, const int* in_sizes, int n_in,
                              void* d_out, int out_size, void* d_ws,
                              size_t ws_size, hipStream_t stream) {
  const float* x       = (const float*)d_in[0];  // [B, F]
  const float* cross_w = (const float*)d_in[1];  // [L, F]
  const float* cross_b = (const float*)d_in[2];  // [L, F]
  const float* W_out   = (const float*)d_in[3];  // [F, T]
  const float* b_out   = (const float*)d_in[4];  // [T]
  float* out = (float*)d_out;                    // [B, T]

  float* bpack = (float*)d_ws;                         // 32 KB
  float* csP   = (float*)((char*)d_ws + 32768);        // 12 floats

  precompute_kernel<<<1, 768, 0, stream>>>(cross_w, cross_b, W_out, b_out, csP);
  pack_b_kernel<<<(KBLKS * 64 + 255) / 256, 256, 0, stream>>>(cross_w, W_out,
                                                              bpack);
  dcn_wmma_kernel<<<B_ROWS / (16 * 8), 256, 0, stream>>>(x, bpack, csP, out);
}